// PermutohedralX_61546881352360
// MI455X (gfx1250) — compile-verified
//
#include <hip/hip_runtime.h>
#include <hip/hip_bf16.h>

// Permutohedral lattice: splat -> 6x blur -> slice.
// N=262144 points, C=64 channels, D1=6 simplex vertices, M=262144 lattice pts.
// vals buffers: (M+1) x 64 floats = 64MB each; the two ping-pong buffers
// (128MB) are kept resident in the 192MB L2. Single-use streams (inp,
// blur_neighbors, out) use non-temporal hints so they don't evict them.

#define PERM_C   64
#define PERM_C4  16      // C / 4 (float4 per row)
#define PERM_D1  6

#if defined(__has_builtin)
#if __has_builtin(__builtin_amdgcn_global_load_async_to_lds_b128)
#define PERM_HAVE_ASYNC 1
#endif
#endif

typedef float v4f __attribute__((ext_vector_type(4)));
#if defined(PERM_HAVE_ASYNC)
typedef int v4i __attribute__((ext_vector_type(4)));
typedef v4i __attribute__((address_space(1)))* perm_gptr_t;   // global v4i*
typedef v4i __attribute__((address_space(3)))* perm_lptr_t;   // LDS v4i*
#endif

// ---------------------------------------------------------------------------
// PROBE (never launched): first function in the file so the disasm snippet
// shows exactly how __hip_atomic_fetch_add(float, relaxed, agent) lowers on
// gfx1250 — native global_atomic_add_f32 vs. a CAS loop.
// ---------------------------------------------------------------------------
__global__ void perm_atomic_probe(float* p, float v) {
    (void)__hip_atomic_fetch_add(p + threadIdx.x, v,
                                 __ATOMIC_RELAXED, __HIP_MEMORY_SCOPE_AGENT);
}

// ---------------------------------------------------------------------------
// Splat: vals[os[n][j]+1][c] += ws[n][j] * inp[n][c]   (atomic scatter-add)
// 4 points per 256-thread block; 64 threads (=2 waves) per point, one channel
// each. ws/os staged through LDS. inp is read exactly once -> non-temporal.
// ---------------------------------------------------------------------------
__global__ void perm_splat_kernel(const float* __restrict__ inp,
                                  const float* __restrict__ ws,
                                  const int*   __restrict__ os,
                                  float*       __restrict__ vals,
                                  int N) {
    __shared__ float s_w[4][PERM_D1];
    __shared__ int   s_o[4][PERM_D1];

    const int tid = threadIdx.x;
    const int blockPoint = blockIdx.x * 4;

    if (tid < 4 * PERM_D1) {
        const int p = tid / PERM_D1;
        const int j = tid % PERM_D1;
        const int n = blockPoint + p;
        if (n < N) {
            s_w[p][j] = ws[(size_t)n * PERM_D1 + j];
            s_o[p][j] = os[(size_t)n * PERM_D1 + j] + 1;   // padded (1-based) row
        }
    }
    __syncthreads();

    const int p = tid >> 6;       // 0..3
    const int c = tid & 63;       // 0..63
    const int n = blockPoint + p;
    if (n >= N) return;

    const float x = __builtin_nontemporal_load(inp + (size_t)n * PERM_C + c);

#pragma unroll
    for (int j = 0; j < PERM_D1; ++j) {
        float* dst = &vals[(size_t)s_o[p][j] * PERM_C + c];
        (void)__hip_atomic_fetch_add(dst, s_w[p][j] * x,
                                     __ATOMIC_RELAXED, __HIP_MEMORY_SCOPE_AGENT);
    }
}

// ---------------------------------------------------------------------------
// One blur pass: dst[m] = src[m] + 0.5*(src[n1] + src[n2]) for m in 1..M,
// with (n1,n2) = blur_neighbors[j][m-1][0..1] (1-based; row 0 == zero pad).
// 16 rows per 256-thread block; 16 threads x float4 cover one 64-float row.
// Streaming diagonal row uses the CDNA5 async global->LDS copy (ASYNCcnt) so
// it overlaps the neighbor-index load and the two gathers; each lane reads
// back only the LDS element it loaded itself, so a per-wave s_wait_asynccnt
// suffices (no cross-lane handoff). Neighbor pairs are read once -> NT i64.
// ---------------------------------------------------------------------------
__global__ void perm_blur_kernel(const float4* __restrict__ src,
                                 float4*       __restrict__ dst,
                                 const long long* __restrict__ bn_j,  // [M] packed (n1,n2)
                                 int M) {
    __shared__ long long s_n[16];
#if defined(PERM_HAVE_ASYNC)
    __shared__ float4 s_diag[16][PERM_C4];
#endif

    const int tid     = threadIdx.x;
    const int ty      = tid >> 4;          // row within block, 0..15
    const int c4      = tid & 15;          // float4 lane within row
    const int rowBase = 1 + blockIdx.x * 16;
    const int m       = rowBase + ty;
    const bool valid  = (m <= M);

#if defined(PERM_HAVE_ASYNC)
    if (valid) {
        // Each lane pulls its own float4 of the diagonal row into LDS.
        __builtin_amdgcn_global_load_async_to_lds_b128(
            (perm_gptr_t)(src + (size_t)m * PERM_C4 + c4),
            (perm_lptr_t)&s_diag[ty][c4],
            0, 0);
    }
#endif

    if (tid < 16) {
        const int mr = rowBase + tid;
        if (mr <= M) s_n[tid] = __builtin_nontemporal_load(bn_j + (mr - 1));
    }
    __syncthreads();
    if (!valid) return;

    const long long nn = s_n[ty];
    const int n1 = (int)(nn & 0xffffffffll);
    const int n2 = (int)(nn >> 32);

    const float4 a = src[(size_t)n1 * PERM_C4 + c4];
    const float4 b = src[(size_t)n2 * PERM_C4 + c4];

#if defined(PERM_HAVE_ASYNC)
    asm volatile("s_wait_asynccnt 0x0" ::: "memory");
    const float4 dg = s_diag[ty][c4];
#else
    const float4 dg = src[(size_t)m * PERM_C4 + c4];
#endif

    float4 r;
    r.x = dg.x + 0.5f * (a.x + b.x);
    r.y = dg.y + 0.5f * (a.y + b.y);
    r.z = dg.z + 0.5f * (a.z + b.z);
    r.w = dg.w + 0.5f * (a.w + b.w);
    dst[(size_t)m * PERM_C4 + c4] = r;
}

// ---------------------------------------------------------------------------
// Slice: out[n][c] = alpha * sum_j ws[n][j] * vals[os[n][j]+1][c]
// 16 points per 256-thread block; 16 threads x float4 per point.
// Output is written once and never re-read -> non-temporal B128 store, so
// the 64MB output stream does not evict the L2-resident vals buffers.
// ---------------------------------------------------------------------------
__global__ void perm_slice_kernel(const float4* __restrict__ vals,
                                  const float*  __restrict__ ws,
                                  const int*    __restrict__ os,
                                  v4f*          __restrict__ out,
                                  int N, float alpha) {
    __shared__ float s_w[16][PERM_D1];
    __shared__ int   s_o[16][PERM_D1];

    const int tid  = threadIdx.x;
    const int base = blockIdx.x * 16;

    if (tid < 16 * PERM_D1) {
        const int p = tid / PERM_D1;
        const int j = tid % PERM_D1;
        const int n = base + p;
        if (n < N) {
            s_w[p][j] = ws[(size_t)n * PERM_D1 + j];
            s_o[p][j] = os[(size_t)n * PERM_D1 + j] + 1;
        }
    }
    __syncthreads();

    const int p  = tid >> 4;
    const int c4 = tid & 15;
    const int n  = base + p;
    if (n >= N) return;

    // Prefetch the 6 gathered rows (gfx1250 global_prefetch_b8), then gather.
#pragma unroll
    for (int j = 0; j < PERM_D1; ++j) {
        __builtin_prefetch(vals + (size_t)s_o[p][j] * PERM_C4 + c4, 0, 0);
    }

    float ax = 0.f, ay = 0.f, az = 0.f, aw = 0.f;
#pragma unroll
    for (int j = 0; j < PERM_D1; ++j) {
        const float  w = s_w[p][j];
        const float4 v = vals[(size_t)s_o[p][j] * PERM_C4 + c4];
        ax += w * v.x; ay += w * v.y; az += w * v.z; aw += w * v.w;
    }
    v4f r;
    r.x = alpha * ax; r.y = alpha * ay; r.z = alpha * az; r.w = alpha * aw;
    __builtin_nontemporal_store(r, out + (size_t)n * PERM_C4 + c4);
}

// ---------------------------------------------------------------------------
// Zero a float4 buffer (graph-replay safe re-init of the accumulator).
// Regular-temporal stores: this buffer is the atomic target and we WANT it
// pulled into / kept in L2.
// ---------------------------------------------------------------------------
__global__ void perm_zero_kernel(float4* __restrict__ buf, unsigned long long n4) {
    unsigned long long i = (unsigned long long)blockIdx.x * blockDim.x + threadIdx.x;
    if (i < n4) {
        float4 z; z.x = 0.f; z.y = 0.f; z.z = 0.f; z.w = 0.f;
        buf[i] = z;
    }
}

// ---------------------------------------------------------------------------
// Host launcher.
// Inputs: d_in[0]=inp [N*C] f32, d_in[1]=ws [N*6] f32, d_in[2]=os [N*6] i32,
//         d_in[3]=blur_neighbors [6*M*2] i32, d_in[4]=M scalar (unused; derived).
// ---------------------------------------------------------------------------
extern "C" void kernel_launch(void* const* d_in, const int* in_sizes, int n_in,
                              void* d_out, int out_size, void* d_ws, size_t ws_size,
                              hipStream_t stream) {
    const float* inp = (const float*)d_in[0];
    const float* ws  = (const float*)d_in[1];
    const int*   os  = (const int*)d_in[2];
    const int*   bn  = (const int*)d_in[3];

    const int N = in_sizes[0] / PERM_C;                 // 262144
    const int M = in_sizes[3] / (2 * PERM_D1);          // 262144
    const int d = PERM_D1 - 1;                          // 5
    const float alpha = 1.0f / (1.0f + exp2f((float)-d));   // 32/33

    const size_t bufElems4 = (size_t)(M + 1) * PERM_C4; // float4 elements/buffer
    float4* valsA = (float4*)d_ws;
    float4* valsB = valsA + bufElems4;                  // needs 2*64MB of d_ws

    // Re-init accumulator A fully + row 0 of B (pad row must read as zero).
    {
        unsigned blocks = (unsigned)((bufElems4 + 255) / 256);
        perm_zero_kernel<<<blocks, 256, 0, stream>>>(valsA, (unsigned long long)bufElems4);
        perm_zero_kernel<<<1, 256, 0, stream>>>(valsB, (unsigned long long)PERM_C4);
    }

    // Splat.
    {
        unsigned blocks = (unsigned)((N + 3) / 4);
        perm_splat_kernel<<<blocks, 256, 0, stream>>>(inp, ws, os, (float*)valsA, N);
    }

    // 6 blur passes, ping-pong A<->B (even count -> final result back in A).
    const float4* src = valsA;
    float4*       dst = valsB;
    const unsigned blurBlocks = (unsigned)((M + 15) / 16);
    for (int j = 0; j < PERM_D1; ++j) {
        const long long* bn_j = (const long long*)(bn + (size_t)j * M * 2);
        perm_blur_kernel<<<blurBlocks, 256, 0, stream>>>(src, dst, bn_j, M);
        const float4* t = src; src = dst; dst = (float4*)t;
    }

    // Slice.
    {
        unsigned blocks = (unsigned)((N + 15) / 16);
        perm_slice_kernel<<<blocks, 256, 0, stream>>>(src, ws, os, (v4f*)d_out, N, alpha);
    }
}